// GenerativeModel_25821343384066
// MI455X (gfx1250) — compile-verified
//
#include <hip/hip_runtime.h>
#include <math.h>

#define TPB     256
#define NPIX    12288          // 3*64*64 floats per sample
#define CHUNKS  12             // NPIX / (TPB*4)
#define LOG_2PI 1.8378770664093453f

__device__ __forceinline__ float sigmoidf(float x) {
    return 1.0f / (1.0f + __expf(-x));
}

__global__ __launch_bounds__(TPB) void genmodel_logprob(
    const float* __restrict__ latent,
    const float* __restrict__ obs,
    const float* __restrict__ raw_size,
    const float* __restrict__ raw_color,
    const float* __restrict__ raw_cs,
    const float* __restrict__ raw_blur,
    float* __restrict__ out)
{
    __shared__ float FX[64];
    __shared__ float FY[64];
    __shared__ float G[192];                       // color[c] * fy[y]
    __shared__ float partial1[TPB / 32];
    __shared__ float partial2[TPB / 32];
    __shared__ __align__(16) float stage[2][TPB * 4];   // 2 x 4KB async staging

    const int n = blockIdx.x;
    const int t = threadIdx.x;

    // ---- uniform scalar parameters ----
    const float size = sigmoidf(raw_size[0]);
    const float k    = logf(1.0f + __expf(raw_cs[0])) /
                       (logf(1.0f + __expf(raw_blur[0])) + 0.001f);
    const float lat  = latent[n];
    const float mx   = sigmoidf(lat) * (1.0f - size);

    // ---- separable occupancy tables (128 sigmoid pairs total per block) ----
    if (t < 64) {
        float gx = ((float)t + 0.5f) * (1.0f / 64.0f);
        FX[t] = sigmoidf(k * (gx - mx)) * sigmoidf(k * (mx + size - gx));
    } else if (t < 128) {
        int y = t - 64;
        float gy = 1.0f - ((float)y + 0.5f) * (1.0f / 64.0f);
        FY[y] = sigmoidf(k * gy) * sigmoidf(k * (size - gy));
    }
    __syncthreads();
    if (t < 192) {
        int c = t >> 6;
        int y = t & 63;
        G[t] = sigmoidf(raw_color[c]) * FY[y];
    }
    __syncthreads();

    // Each thread handles float4 chunk (t + 256*j): x-phase constant over j,
    // row index = (t>>4) + 16*j  (chunks never cross a 64-float row).
    const int    x0  = (t & 15) * 4;
    const float4 fx4 = *(const float4*)&FX[x0];
    float gf[CHUNKS];
#pragma unroll
    for (int j = 0; j < CHUNKS; ++j) gf[j] = G[(t >> 4) + 16 * j];

    // ---- async global -> LDS double-buffered streaming of the 48KB slab ----
    const float*  sbase = obs + (size_t)n * NPIX;          // uniform per block
    const unsigned voff = (unsigned)(t * 16);
    const unsigned lds0 = (unsigned)(uintptr_t)(&stage[0][t * 4]);
    const unsigned lds1 = (unsigned)(uintptr_t)(&stage[1][t * 4]);

#define ISSUE(J)                                                              \
    do {                                                                      \
        unsigned _l = (((J) & 1) ? lds1 : lds0);                              \
        unsigned _v = voff + (unsigned)((J) * (TPB * 16));                    \
        asm volatile("global_load_async_to_lds_b128 %0, %1, %2"               \
                     :: "v"(_l), "v"(_v), "s"(sbase) : "memory");             \
    } while (0)

    float acc1 = 0.0f;   // sum obs * fx * (color*fy)
    float acc2 = 0.0f;   // sum obs^2

    ISSUE(0);
    ISSUE(1);
#pragma unroll
    for (int j = 0; j < CHUNKS; ++j) {
        if (j < CHUNKS - 1) asm volatile("s_wait_asynccnt 0x1" ::: "memory");
        else                asm volatile("s_wait_asynccnt 0x0" ::: "memory");

        const float4 o = *(const float4*)&stage[j & 1][t * 4];

        acc2 = fmaf(o.x, o.x, fmaf(o.y, o.y, fmaf(o.z, o.z, fmaf(o.w, o.w, acc2))));
        float s = o.x * fx4.x;
        s = fmaf(o.y, fx4.y, s);
        s = fmaf(o.z, fx4.z, s);
        s = fmaf(o.w, fx4.w, s);
        acc1 = fmaf(s, gf[j], acc1);

        if (j + 2 < CHUNKS) {
            // ensure our ds reads of this buffer completed before the async
            // engine can overwrite it
            asm volatile("s_wait_dscnt 0x0" ::: "memory");
            ISSUE(j + 2);
        }
    }
#undef ISSUE

    // ---- reduction: wave32 shfl, then across the 8 waves via LDS ----
#pragma unroll
    for (int off = 16; off > 0; off >>= 1) {
        acc1 += __shfl_down(acc1, off, 32);
        acc2 += __shfl_down(acc2, off, 32);
    }
    const int wave = t >> 5;
    const int lane = t & 31;
    if (lane == 0) { partial1[wave] = acc1; partial2[wave] = acc2; }
    __syncthreads();

    if (t == 0) {
        float a1 = 0.0f, a2 = 0.0f;
#pragma unroll
        for (int w = 0; w < TPB / 32; ++w) { a1 += partial1[w]; a2 += partial2[w]; }

        // closed-form sum loc^2 = (sum c^2)(sum fx^2)(sum fy^2)
        float sfx2 = 0.0f, sfy2 = 0.0f;
        for (int i = 0; i < 64; ++i) {
            sfx2 = fmaf(FX[i], FX[i], sfx2);
            sfy2 = fmaf(FY[i], FY[i], sfy2);
        }
        float sc2 = 0.0f;
        for (int c = 0; c < 3; ++c) {
            float cc = sigmoidf(raw_color[c]);
            sc2 = fmaf(cc, cc, sc2);
        }
        float loc2 = sc2 * sfx2 * sfy2;

        // out = latent prior + obs log-prob
        float res = -0.5f * lat * lat - 0.5f * LOG_2PI
                  + a1 - 0.5f * a2 - 0.5f * loc2
                  - (float)NPIX * 0.5f * LOG_2PI;
        out[n] = res;
    }
}

extern "C" void kernel_launch(void* const* d_in, const int* in_sizes, int n_in,
                              void* d_out, int out_size, void* d_ws, size_t ws_size,
                              hipStream_t stream) {
    const float* latent   = (const float*)d_in[0];
    const float* obs      = (const float*)d_in[1];
    const float* raw_size = (const float*)d_in[2];
    const float* raw_col  = (const float*)d_in[3];
    const float* raw_cs   = (const float*)d_in[4];
    const float* raw_blur = (const float*)d_in[5];
    float* out = (float*)d_out;

    const int n_samples = in_sizes[0];   // 64*128 = 8192
    genmodel_logprob<<<n_samples, TPB, 0, stream>>>(
        latent, obs, raw_size, raw_col, raw_cs, raw_blur, out);
}